// Torch_Ops_Aten_GridSampler2DBackward_OutModule_53987738910878
// MI455X (gfx1250) — compile-verified
//
#include <hip/hip_runtime.h>

// grid_sampler_2d backward (bilinear, zeros padding) for
//   grad_output [8,256,96,96], input [8,256,96,96], grid [8,96,96,2]
// Outputs concatenated in d_out: grad_input (18,874,368 f32) then grad_grid (147,456 f32).
//
// Memory-bound scatter/gather op: no matmul structure -> WMMA not applicable.
// CDNA5 paths used:
//   - global_load_async_to_lds_b128 + s_wait_asynccnt (tile grid staging, ASYNCcnt path)
//   - global_prefetch (strided grad_output stream)
//   - hardware non-returning global_atomic_add_f32 scatter
//   - NT temporal hints on the streaming grad_output reads / grad_grid writes so the
//     192 MB L2 keeps the 151 MB input+grad_input gather/scatter set resident.

#define NBATCH 8
#define NCH    256
#define HIN    96
#define WIN    96
#define HOUT   96
#define WOUT   96
#define PLANE  (HIN * WIN)            // 9216 (== HOUT*WOUT)
#define NPIX   (NBATCH * HOUT * WOUT) // 73728
#define CSPLIT 4
#define CCHUNK (NCH / CSPLIT)         // 64 channels per thread
#define PIXPB  64                     // pixels per block (blockDim.x)

typedef float v2f __attribute__((ext_vector_type(2))); // native vector for NT builtins

__global__ __launch_bounds__(256) void gs2d_bwd_kernel(
    const float* __restrict__ grad_out,
    const float* __restrict__ inp,
    const float* __restrict__ grid,
    const int*   __restrict__ ac_flag,
    float* __restrict__ grad_inp,
    float* __restrict__ grad_grid)
{
    __shared__ __align__(16) float sgrid[PIXPB * 2]; // 512 B: this tile's (gx,gy)
    __shared__ float redx[CSPLIT][PIXPB];
    __shared__ float redy[CSPLIT][PIXPB];

    const int x = threadIdx.x;      // pixel within tile
    const int y = threadIdx.y;      // channel chunk
    const int pixbase = blockIdx.x * PIXPB;

    // ---- Wave 0 stages this tile's grid coords into LDS via async copy ----
    // 32 lanes x 16 B = 512 B. flat ids 0..31 == exactly wave 0 (uniform branch).
    if (y == 0 && x < 32) {
        // Flat pointer to __shared__ truncates to the wave-relative LDS byte
        // address (flat->LDS mapping keeps addr[31:0]).
        unsigned     lds_addr = (unsigned)(size_t)(sgrid + x * 4);
        const float* gsrc     = grid + (size_t)pixbase * 2 + x * 4;
        asm volatile("global_load_async_to_lds_b128 %0, %1, off"
                     :: "v"(lds_addr), "v"(gsrc) : "memory");
        asm volatile("s_wait_asynccnt 0" ::: "memory");
    }
    __syncthreads();

    const int  pix = pixbase + x;
    const int  n   = pix / PLANE;
    const int  sp  = pix - n * PLANE;
    const bool ac  = (*ac_flag) != 0;

    const v2f g2 = ((const v2f*)sgrid)[x];
    const float gx = g2.x, gy = g2.y;

    const float ix = ac ? (gx + 1.0f) * 0.5f * (float)(WIN - 1)
                        : ((gx + 1.0f) * (float)WIN - 1.0f) * 0.5f;
    const float iy = ac ? (gy + 1.0f) * 0.5f * (float)(HIN - 1)
                        : ((gy + 1.0f) * (float)HIN - 1.0f) * 0.5f;

    const float x0f = floorf(ix), y0f = floorf(iy);
    const float wx1 = ix - x0f, wx0 = 1.0f - wx1;
    const float wy1 = iy - y0f, wy0 = 1.0f - wy1;

    const int x0 = (int)x0f, y0 = (int)y0f;
    const int x1 = x0 + 1,  y1 = y0 + 1;

    const bool vx0 = (x0 >= 0) & (x0 < WIN);
    const bool vx1 = (x1 >= 0) & (x1 < WIN);
    const bool vy0 = (y0 >= 0) & (y0 < HIN);
    const bool vy1 = (y1 >= 0) & (y1 < HIN);
    const float m00 = (vx0 & vy0) ? 1.0f : 0.0f;
    const float m01 = (vx1 & vy0) ? 1.0f : 0.0f;
    const float m10 = (vx0 & vy1) ? 1.0f : 0.0f;
    const float m11 = (vx1 & vy1) ? 1.0f : 0.0f;

    const int x0c = min(max(x0, 0), WIN - 1), x1c = min(max(x1, 0), WIN - 1);
    const int y0c = min(max(y0, 0), HIN - 1), y1c = min(max(y1, 0), HIN - 1);

    const int off00 = y0c * WIN + x0c, off01 = y0c * WIN + x1c;
    const int off10 = y1c * WIN + x0c, off11 = y1c * WIN + x1c;

    // Scatter weights with validity folded in (invalid -> adds exact 0.0 at a
    // clamped, always-in-bounds address; matches the JAX vjp structurally).
    const float s00 = wy0 * wx0 * m00, s01 = wy0 * wx1 * m01;
    const float s10 = wy1 * wx0 * m10, s11 = wy1 * wx1 * m11;

    const size_t base = (size_t)(n * NCH + y * CCHUNK) * PLANE;
    const float* __restrict__ gop   = grad_out + base + sp;
    const float* __restrict__ inp_b = inp      + base;
    float*       __restrict__ gi_b  = grad_inp + base;

    float gix = 0.0f, giy = 0.0f;

    #pragma unroll 8
    for (int c = 0; c < CCHUNK; ++c) {
        const int co = c * PLANE;
        if (c + 16 < CCHUNK)
            __builtin_prefetch(gop + (size_t)(c + 16) * PLANE, 0, 1); // global_prefetch

        // Streaming, read-once operand: NT hint keeps it from evicting the
        // L2-resident input/grad_input planes.
        const float g   = __builtin_nontemporal_load(gop + co);
        const float a00 = inp_b[co + off00] * m00; // L2-resident random gathers
        const float a01 = inp_b[co + off01] * m01;
        const float a10 = inp_b[co + off10] * m10;
        const float a11 = inp_b[co + off11] * m11;

        gix = fmaf(g, wy0 * (a01 - a00) + wy1 * (a11 - a10), gix);
        giy = fmaf(g, wx0 * (a10 - a00) + wx1 * (a11 - a01), giy);

        atomicAdd(gi_b + co + off00, s00 * g);     // non-returning global_atomic_add_f32
        atomicAdd(gi_b + co + off01, s01 * g);
        atomicAdd(gi_b + co + off10, s10 * g);
        atomicAdd(gi_b + co + off11, s11 * g);
    }

    // Reduce the 4 channel-chunk partials per pixel through LDS.
    redx[y][x] = gix;
    redy[y][x] = giy;
    __syncthreads();

    if (y == 0) {
        const float sx = redx[0][x] + redx[1][x] + redx[2][x] + redx[3][x];
        const float sy = redy[0][x] + redy[1][x] + redy[2][x] + redy[3][x];
        const float mx = ac ? 0.5f * (float)(WIN - 1) : 0.5f * (float)WIN;
        const float my = ac ? 0.5f * (float)(HIN - 1) : 0.5f * (float)HIN;
        v2f o;
        o.x = sx * mx;   // d/d gx
        o.y = sy * my;   // d/d gy
        // Write-once result, never re-read by this dispatch: NT store.
        __builtin_nontemporal_store(o, (v2f*)grad_grid + pix);
    }
}

// Zero the full output buffer (grad_input needs zero-init for the atomic scatter;
// harness poisons d_out with 0xAA before timing). Default (RT) stores on purpose:
// these lines are immediately re-hit by the atomic scatter, so keep them in L2.
__global__ void gs2d_zero_kernel(float* __restrict__ p, int n)
{
    const int n4 = n >> 2;
    float4* p4 = (float4*)p;
    const float4 z = {0.0f, 0.0f, 0.0f, 0.0f};
    int i = blockIdx.x * blockDim.x + threadIdx.x;
    const int stride = gridDim.x * blockDim.x;
    for (; i < n4; i += stride) p4[i] = z;
    // tail (n not multiple of 4)
    const int tail = n & 3;
    if (blockIdx.x == 0 && threadIdx.x < tail) p[n4 * 4 + threadIdx.x] = 0.0f;
}

extern "C" void kernel_launch(void* const* d_in, const int* in_sizes, int n_in,
                              void* d_out, int out_size, void* d_ws, size_t ws_size,
                              hipStream_t stream)
{
    const float* grad_out = (const float*)d_in[0];
    const float* inp      = (const float*)d_in[1];
    const float* grid     = (const float*)d_in[2];
    const int*   ac_flag  = (const int*)  d_in[5];

    float* out       = (float*)d_out;
    float* grad_inp  = out;
    float* grad_grid = out + (size_t)NBATCH * NCH * HIN * WIN;

    gs2d_zero_kernel<<<2048, 256, 0, stream>>>(out, out_size);

    dim3 blk(PIXPB, CSPLIT, 1);        // 256 threads = 8 wave32
    dim3 grd(NPIX / PIXPB, 1, 1);      // 1152 blocks
    gs2d_bwd_kernel<<<grd, blk, 0, stream>>>(grad_out, inp, grid, ac_flag,
                                             grad_inp, grad_grid);
}